// ContentAwareUpsampling_80917183857371
// MI455X (gfx1250) — compile-verified
//
#include <hip/hip_runtime.h>
#include <hip/hip_bf16.h>
#include <math.h>

#define BN_EPS 1e-5f

typedef __attribute__((ext_vector_type(2))) float v2f;
typedef __attribute__((ext_vector_type(8))) float v8f;
typedef __attribute__((ext_vector_type(4))) unsigned int u32x4;
typedef __attribute__((ext_vector_type(8))) int i32x8;
typedef __attribute__((ext_vector_type(4))) int i32x4;

#if __has_builtin(__builtin_amdgcn_tensor_load_to_lds)
#define USE_TDM 1
#endif

// Problem constants
#define NB   2
#define CIN  256
#define CC   64
#define HH   48
#define WW   48
#define PIX  (HH*WW)          // 2304
#define KUP  100              // UP*UP*K*K
#define OH   96
#define OW   96
#define FH   192
#define FW   192
#define FP   (FH*FW)          // 36864
#define NT5  (FP/16)          // 2304 column tiles in k5

// ---------------------------------------------------------------------------
// K1: compress 1x1 conv  ==  GEMM  (64 x 256) * (256 x 2304)  + BN + ReLU
// one wave computes a 64x16 output tile (4 WMMA accumulators share B)
// grid: 2 batches * 144 col-tiles = 288 waves = 72 blocks * 128 threads
// ---------------------------------------------------------------------------
__global__ __launch_bounds__(128)
void k1_compress(const float* __restrict__ x, const float* __restrict__ w,
                 const float* __restrict__ g, const float* __restrict__ bb,
                 const float* __restrict__ mm, const float* __restrict__ vv,
                 float* __restrict__ comp)
{
    const int wave = (blockIdx.x * blockDim.x + threadIdx.x) >> 5;
    const int lane = threadIdx.x & 31;
    const int ntiles = PIX / 16;            // 144
    const int n  = wave / ntiles;           // batch
    const int nt = wave % ntiles;

    const float* B = x + (size_t)n * CIN * PIX;
    const int col  = nt * 16 + (lane & 15);
    const int kb   = (lane >> 4) * 2;
    const int mrow = lane & 15;

    v8f acc[4] = {};
    for (int k = 0; k < CIN; k += 4) {
        v2f bv;
        bv.x = B[(size_t)(k + kb) * PIX + col];
        bv.y = B[(size_t)(k + kb + 1) * PIX + col];
#pragma unroll
        for (int mt = 0; mt < 4; ++mt) {
            const float* ar = w + (size_t)(mt * 16 + mrow) * CIN + k + kb;
            v2f av; av.x = ar[0]; av.y = ar[1];
            acc[mt] = __builtin_amdgcn_wmma_f32_16x16x4_f32(
                false, av, false, bv, (short)0, acc[mt], false, false);
        }
    }
#pragma unroll
    for (int mt = 0; mt < 4; ++mt) {
#pragma unroll
        for (int r = 0; r < 8; ++r) {
            const int cc  = mt * 16 + (lane >> 4) * 8 + r;
            const float inv = g[cc] * rsqrtf(vv[cc] + BN_EPS);
            const float bet = bb[cc] - mm[cc] * inv;
            const float y   = fmaf(acc[mt][r], inv, bet);
            comp[(size_t)n * CC * PIX + (size_t)cc * PIX + col] = fmaxf(y, 0.f);
        }
    }
}

// ---------------------------------------------------------------------------
// K2: ke1 3x3 dilated conv (dil=2, pad=2) 64->64 + BN + ReLU   (direct VALU)
// ---------------------------------------------------------------------------
__global__ __launch_bounds__(256)
void k2_ke1(const float* __restrict__ comp, const float* __restrict__ w,
            const float* __restrict__ g, const float* __restrict__ bb,
            const float* __restrict__ mm, const float* __restrict__ vv,
            float* __restrict__ out)
{
    const int idx = blockIdx.x * 256 + threadIdx.x;
    const int wc = idx % WW;
    const int h  = (idx / WW) % HH;
    const int co = (idx / PIX) % CC;
    const int n  = idx / (CC * PIX);

    const float* inB = comp + (size_t)n * CC * PIX;
    float acc = 0.f;
    for (int kh = 0; kh < 3; ++kh) {
        const int y = h + 2 * (kh - 1);
        if (y < 0 || y >= HH) continue;
        for (int kw = 0; kw < 3; ++kw) {
            const int xx = wc + 2 * (kw - 1);
            if (xx < 0 || xx >= WW) continue;
            const float* ip = inB + y * WW + xx;
            const float* wp = w + (size_t)co * (CC * 9) + kh * 3 + kw;
#pragma unroll 8
            for (int ci = 0; ci < CC; ++ci)
                acc = fmaf(ip[(size_t)ci * PIX], wp[ci * 9], acc);
        }
    }
    const float inv = g[co] * rsqrtf(vv[co] + BN_EPS);
    const float bet = bb[co] - mm[co] * inv;
    out[idx] = fmaxf(fmaf(acc, inv, bet), 0.f);
}

// ---------------------------------------------------------------------------
// K3a: ke2 3x3 conv (pad=1) 64->100 -> logits
// ---------------------------------------------------------------------------
__global__ __launch_bounds__(256)
void k3a_ke2(const float* __restrict__ k1, const float* __restrict__ w,
             float* __restrict__ klog)
{
    const int idx = blockIdx.x * 256 + threadIdx.x;
    const int wc = idx % WW;
    const int h  = (idx / WW) % HH;
    const int co = (idx / PIX) % KUP;
    const int n  = idx / (KUP * PIX);

    const float* inB = k1 + (size_t)n * CC * PIX;
    float acc = 0.f;
    for (int kh = 0; kh < 3; ++kh) {
        const int y = h + kh - 1;
        if (y < 0 || y >= HH) continue;
        for (int kw = 0; kw < 3; ++kw) {
            const int xx = wc + kw - 1;
            if (xx < 0 || xx >= WW) continue;
            const float* ip = inB + y * WW + xx;
            const float* wp = w + (size_t)co * (CC * 9) + kh * 3 + kw;
#pragma unroll 8
            for (int ci = 0; ci < CC; ++ci)
                acc = fmaf(ip[(size_t)ci * PIX], wp[ci * 9], acc);
        }
    }
    klog[idx] = acc;
}

// ---------------------------------------------------------------------------
// K3b: in-place softmax over the 25 taps, per (n, u, h, w)
// ---------------------------------------------------------------------------
__global__ __launch_bounds__(256)
void k3b_softmax(float* __restrict__ klog)
{
    const int idx = blockIdx.x * 256 + threadIdx.x;
    const int wc = idx % WW;
    const int h  = (idx / WW) % HH;
    const int u  = (idx / PIX) % 4;
    const int n  = idx / (4 * PIX);

    float* base = klog + (size_t)n * KUP * PIX + (size_t)u * 25 * PIX + h * WW + wc;
    float mx = -3.4e38f;
#pragma unroll
    for (int k = 0; k < 25; ++k) mx = fmaxf(mx, base[(size_t)k * PIX]);
    float s = 0.f;
    float e[25];
#pragma unroll
    for (int k = 0; k < 25; ++k) { e[k] = __expf(base[(size_t)k * PIX] - mx); s += e[k]; }
    const float rs = 1.f / s;
#pragma unroll
    for (int k = 0; k < 25; ++k) base[(size_t)k * PIX] = e[k] * rs;
}

// ---------------------------------------------------------------------------
// K4: content reassembly, fused bilinear-upsample of kernels (LDS-staged) +
// nearest-2x unfold of x + 25-tap dot, scatter-stored in pixel-shuffled layout
// ---------------------------------------------------------------------------
__global__ __launch_bounds__(256)
void k4_reassemble(const float* __restrict__ x, const float* __restrict__ ksm,
                   float* __restrict__ shuf)
{
    __shared__ float wk[KUP];
    const int b  = blockIdx.x;
    const int ow = b % OW;
    const int oh = (b / OW) % OH;
    const int n  = b / (OH * OW);
    const int c  = threadIdx.x;

    if (c < KUP) {
        float sy = oh * 0.5f - 0.25f;  sy = fminf(fmaxf(sy, 0.f), (float)(HH - 1));
        float sx = ow * 0.5f - 0.25f;  sx = fminf(fmaxf(sx, 0.f), (float)(WW - 1));
        const int y0 = (int)floorf(sy), x0 = (int)floorf(sx);
        const int y1 = min(y0 + 1, HH - 1), x1 = min(x0 + 1, WW - 1);
        const float wy = sy - y0, wx = sx - x0;
        const float* kb = ksm + (size_t)n * KUP * PIX + (size_t)c * PIX;
        const float v00 = kb[y0 * WW + x0], v01 = kb[y0 * WW + x1];
        const float v10 = kb[y1 * WW + x0], v11 = kb[y1 * WW + x1];
        wk[c] = (v00 * (1.f - wx) + v01 * wx) * (1.f - wy)
              + (v10 * (1.f - wx) + v11 * wx) * wy;
    }
    __syncthreads();

    const float* xb = x + ((size_t)n * CIN + c) * PIX;
    float acc[4] = {0.f, 0.f, 0.f, 0.f};
#pragma unroll
    for (int kh = 0; kh < 5; ++kh) {
        const int ys = oh + kh - 2;
        if (ys < 0 || ys >= OH) continue;
        const int iy = ys >> 1;
#pragma unroll
        for (int kw = 0; kw < 5; ++kw) {
            const int xs = ow + kw - 2;
            if (xs < 0 || xs >= OW) continue;
            const float xv = xb[iy * WW + (xs >> 1)];
            const int k = kh * 5 + kw;
#pragma unroll
            for (int u = 0; u < 4; ++u)
                acc[u] = fmaf(wk[u * 25 + k], xv, acc[u]);
        }
    }
    const int yy = 2 * oh + ((c >> 1) & 1);
    const int xx = 2 * ow + (c & 1);
#pragma unroll
    for (int u = 0; u < 4; ++u) {
        const int c2 = u * 64 + (c >> 2);
        shuf[((size_t)n * CIN + c2) * FP + (size_t)yy * FW + xx] = acc[u];
    }
}

// ---------------------------------------------------------------------------
// K5: proj 1x1 conv == GEMM (256 x 256) * (256 x 36864) per batch + BN + ReLU
// One block = one 16-column B panel (256x16 f32 = 16 KB LDS), staged ONCE by
// the Tensor Data Mover (2D tile descriptor, tracked via TENSORcnt), then the
// 4 waves of the block cover M = 4 x 64 rows from the shared panel.
// grid: 2 * 2304 = 4608 blocks * 128 threads  -> shuf is read exactly once.
// ---------------------------------------------------------------------------
__global__ __launch_bounds__(128)
void k5_proj(const float* __restrict__ shuf, const float* __restrict__ w,
             const float* __restrict__ g, const float* __restrict__ bb,
             const float* __restrict__ mm, const float* __restrict__ vv,
             float* __restrict__ out)
{
    __shared__ float bpanel[CIN * 16];      // sole LDS object -> offset 0

    const int n  = blockIdx.x / NT5;        // batch
    const int nt = blockIdx.x % NT5;        // 16-column tile
    const float* Bsrc = shuf + (size_t)n * CIN * FP + (size_t)nt * 16;

#ifdef USE_TDM
    if (threadIdx.x < 32) {                 // wave 0 issues the TDM descriptor
        const unsigned long long ga = (unsigned long long)(const void*)Bsrc;
        u32x4 g0;
        g0[0] = 1u;                                    // count=1, user mode
        g0[1] = 0u;                                    // lds_addr = 0 (bpanel)
        g0[2] = (unsigned)(ga & 0xffffffffull);        // global_addr[31:0]
        g0[3] = (unsigned)((ga >> 32) & 0x1ffffffull)  // global_addr[56:32]
              | (2u << 30);                            // type = 2 (image)
        i32x8 g1;
        g1[0] = (int)(2u << 16);                       // data_size = 4 bytes
        g1[1] = (int)(((unsigned)FP & 0xffffu) << 16); // tensor_dim0[15:0]
        g1[2] = (int)((((unsigned)FP >> 16) & 0xffffu)
              | (((unsigned)CIN & 0xffffu) << 16));    // dim0 hi | tensor_dim1 lo
        g1[3] = (int)((((unsigned)CIN >> 16) & 0xffffu)
              | (16u << 16));                          // dim1 hi | tile_dim0 = 16
        g1[4] = (int)(256u);                           // tile_dim1=256, tile_dim2=0
        g1[5] = (int)FP;                               // tensor_dim0_stride[31:0]
        g1[6] = 0;                                     // stride hi | dim1_stride lo
        g1[7] = 0;
        i32x4 g2 = {0, 0, 0, 0};                       // 2D tensor: groups 2/3 unused
        i32x4 g3 = {0, 0, 0, 0};
        i32x8 g4 = {0, 0, 0, 0, 0, 0, 0, 0};           // extra group (clang-23 form)
        __builtin_amdgcn_tensor_load_to_lds(g0, g1, g2, g3, g4, 0);
        __builtin_amdgcn_s_wait_tensorcnt(0);
    }
#else
    // cooperative fallback: 128 threads stage the panel (coalesced 64B rows)
    {
        const int colT = threadIdx.x & 15;
        const int kT   = threadIdx.x >> 4;  // 8 row phases
#pragma unroll
        for (int i = 0; i < 32; ++i) {
            const int k = kT + i * 8;
            bpanel[k * 16 + colT] = Bsrc[(size_t)k * FP + colT];
        }
    }
#endif
    __syncthreads();

    const int mb   = threadIdx.x >> 5;      // wave id = 64-row block 0..3
    const int lane = threadIdx.x & 31;
    const int colL = lane & 15;
    const int col  = nt * 16 + colL;
    const int kb   = (lane >> 4) * 2;
    const int mrow = lane & 15;

    v8f acc[4] = {};
    for (int k = 0; k < CIN; k += 4) {
        v2f bv;                              // conflict-free ds reads
        bv.x = bpanel[(k + kb) * 16 + colL];
        bv.y = bpanel[(k + kb + 1) * 16 + colL];
#pragma unroll
        for (int mt = 0; mt < 4; ++mt) {
            const float* ar = w + (size_t)(mb * 64 + mt * 16 + mrow) * CIN + k + kb;
            v2f av; av.x = ar[0]; av.y = ar[1];
            acc[mt] = __builtin_amdgcn_wmma_f32_16x16x4_f32(
                false, av, false, bv, (short)0, acc[mt], false, false);
        }
    }
#pragma unroll
    for (int mt = 0; mt < 4; ++mt) {
#pragma unroll
        for (int r = 0; r < 8; ++r) {
            const int co  = mb * 64 + mt * 16 + (lane >> 4) * 8 + r;
            const float inv = g[co] * rsqrtf(vv[co] + BN_EPS);
            const float bet = bb[co] - mm[co] * inv;
            const float y   = fmaf(acc[mt][r], inv, bet);
            out[(size_t)n * CIN * FP + (size_t)co * FP + col] = fmaxf(y, 0.f);
        }
    }
}

// ---------------------------------------------------------------------------
extern "C" void kernel_launch(void* const* d_in, const int* in_sizes, int n_in,
                              void* d_out, int out_size, void* d_ws, size_t ws_size,
                              hipStream_t stream)
{
    const float* x          = (const float*)d_in[0];
    const float* compress_w = (const float*)d_in[1];
    const float* g1 = (const float*)d_in[2];
    const float* b1 = (const float*)d_in[3];
    const float* m1 = (const float*)d_in[4];
    const float* v1 = (const float*)d_in[5];
    const float* ke1_w = (const float*)d_in[6];
    const float* gk = (const float*)d_in[7];
    const float* bk = (const float*)d_in[8];
    const float* mk = (const float*)d_in[9];
    const float* vk = (const float*)d_in[10];
    const float* ke2_w  = (const float*)d_in[11];
    const float* proj_w = (const float*)d_in[12];
    const float* g2 = (const float*)d_in[13];
    const float* b2 = (const float*)d_in[14];
    const float* m2 = (const float*)d_in[15];
    const float* v2 = (const float*)d_in[16];
    float* out = (float*)d_out;

    float* ws   = (float*)d_ws;
    float* comp = ws;                                   // 2*64*2304   = 294912
    float* k1b  = comp + (size_t)NB * CC * PIX;         // 294912
    float* klog = k1b  + (size_t)NB * CC * PIX;         // 2*100*2304  = 460800
    float* shuf = klog + (size_t)NB * KUP * PIX;        // 2*256*36864 = 18874368

    k1_compress<<<72, 128, 0, stream>>>(x, compress_w, g1, b1, m1, v1, comp);
    k2_ke1<<<(NB * CC * PIX) / 256, 256, 0, stream>>>(comp, ke1_w, gk, bk, mk, vk, k1b);
    k3a_ke2<<<(NB * KUP * PIX) / 256, 256, 0, stream>>>(k1b, ke2_w, klog);
    k3b_softmax<<<(NB * 4 * PIX) / 256, 256, 0, stream>>>(klog);
    k4_reassemble<<<NB * OH * OW, 256, 0, stream>>>(x, klog, shuf);
    k5_proj<<<NB * NT5, 128, 0, stream>>>(shuf, proj_w, g2, b2, m2, v2, out);

    (void)in_sizes; (void)n_in; (void)out_size; (void)ws_size;
}